// SSIMLoss_25563645346673
// MI455X (gfx1250) — compile-verified
//
#include <hip/hip_runtime.h>

// SSIM loss, B=64 C=2 H=W=320, 7x7 uniform window, VALID -> 314x314, scalar out.
// Bandwidth-bound: 105 MB compulsory reads, ~4.5us floor at 23.3 TB/s.
// Sliding separable box sums (add row i+6 / subtract row i); the re-read for
// the subtract hits the 192MB L2 (both tensors fit), so HBM traffic ~= compulsory.
// Final deterministic reduction uses V_WMMA_F32_16X16X4_F32.

#define WIN      7
#define IMG_H    320
#define IMG_W    320
#define OUT_H    314
#define OUT_W    314
#define BATCH    64
#define TROWS    16
#define NTILES   20                 // ceil(314/16)
#define NPARTIAL (BATCH * NTILES)   // 1280 = 20 chunks of 64

typedef __attribute__((ext_vector_type(2))) float v2f;
typedef __attribute__((ext_vector_type(8))) float v8f;

__global__ __launch_bounds__(IMG_W)
void ssim_partial_kernel(const float* __restrict__ X,
                         const float* __restrict__ Y,
                         const float* __restrict__ drange,
                         float* __restrict__ partials)
{
    __shared__ float sh_x[IMG_W], sh_y[IMG_W], sh_xx[IMG_W], sh_yy[IMG_W], sh_xy[IMG_W];

    const int j    = threadIdx.x;          // column 0..319
    const int b    = blockIdx.x;           // batch image
    const int tile = blockIdx.y;           // row tile
    const int i0   = tile * TROWS;
    const int rows = min(TROWS, OUT_H - i0);

    const float d  = drange[b];
    const float C1 = (0.01f * d) * (0.01f * d);
    const float C2 = (0.03f * d) * (0.03f * d);
    const float inv_np   = 1.0f / 49.0f;   // conv weight 1/(7*7)
    const float cov_norm = 49.0f / 48.0f;  // NP/(NP-1)

    const size_t base0 = ((size_t)b * 2 + 0) * (size_t)IMG_H * IMG_W;
    const size_t base1 = ((size_t)b * 2 + 1) * (size_t)IMG_H * IMG_W;
    const float* X0 = X + base0 + j;
    const float* X1 = X + base1 + j;
    const float* Y0 = Y + base0 + j;
    const float* Y1 = Y + base1 + j;

    // Running 7-row column sums over both channels.
    float cs_x = 0.f, cs_y = 0.f, cs_xx = 0.f, cs_yy = 0.f, cs_xy = 0.f;

    // Prime with the first 6 rows of the window.
    for (int r = i0; r < i0 + WIN - 1; ++r) {
        const size_t off = (size_t)r * IMG_W;
        float x0 = X0[off], x1 = X1[off];
        float y0 = Y0[off], y1 = Y1[off];
        cs_x  += x0 + x1;          cs_y  += y0 + y1;
        cs_xx += x0*x0 + x1*x1;    cs_yy += y0*y0 + y1*y1;
        cs_xy += x0*y0 + x1*y1;
    }

    float acc = 0.0f;
    for (int ii = 0; ii < rows; ++ii) {
        const int i = i0 + ii;

        // Add incoming row i+6.
        const size_t ra = (size_t)(i + WIN - 1) * IMG_W;
        float x0 = X0[ra], x1 = X1[ra];
        float y0 = Y0[ra], y1 = Y1[ra];
        if (ii + 1 < rows) {  // gfx1250 global_prefetch_b8 for the next row
            __builtin_prefetch(X0 + ra + IMG_W, 0, 0);
            __builtin_prefetch(Y0 + ra + IMG_W, 0, 0);
        }
        cs_x  += x0 + x1;          cs_y  += y0 + y1;
        cs_xx += x0*x0 + x1*x1;    cs_yy += y0*y0 + y1*y1;
        cs_xy += x0*y0 + x1*y1;

        sh_x[j]  = cs_x;  sh_y[j]  = cs_y;
        sh_xx[j] = cs_xx; sh_yy[j] = cs_yy; sh_xy[j] = cs_xy;
        __syncthreads();

        if (j < OUT_W) {
            float sx = 0.f, sy = 0.f, sxx = 0.f, syy = 0.f, sxy = 0.f;
#pragma unroll
            for (int t = 0; t < WIN; ++t) {
                sx  += sh_x [j + t];
                sy  += sh_y [j + t];
                sxx += sh_xx[j + t];
                syy += sh_yy[j + t];
                sxy += sh_xy[j + t];
            }
            float ux  = sx  * inv_np, uy  = sy  * inv_np;
            float uxx = sxx * inv_np, uyy = syy * inv_np, uxy = sxy * inv_np;
            float vx  = cov_norm * (uxx - ux * ux);
            float vy  = cov_norm * (uyy - uy * uy);
            float vxy = cov_norm * (uxy - ux * uy);
            float A1  = 2.0f * ux * uy + C1;
            float A2  = 2.0f * vxy + C2;
            float B1  = ux * ux + uy * uy + C1;
            float B2  = vx + vy + C2;
            acc += (A1 * A2) / (B1 * B2);
        }
        __syncthreads();

        // Slide: subtract outgoing row i (re-read is L2-resident).
        const size_t rs = (size_t)i * IMG_W;
        float ox0 = X0[rs], ox1 = X1[rs];
        float oy0 = Y0[rs], oy1 = Y1[rs];
        cs_x  -= ox0 + ox1;            cs_y  -= oy0 + oy1;
        cs_xx -= ox0*ox0 + ox1*ox1;    cs_yy -= oy0*oy0 + oy1*oy1;
        cs_xy -= ox0*oy0 + ox1*oy1;
    }

    // Fixed-order block reduction of 320 per-thread accumulators (deterministic).
    sh_x[j] = acc;
    __syncthreads();
    for (int s = 160; ; s >>= 1) {       // 160,80,40,20,10,5
        if (j < s) sh_x[j] += sh_x[j + s];
        __syncthreads();
        if (s == 5) break;
    }
    if (j == 0) {
        partials[b * NTILES + tile] =
            sh_x[0] + sh_x[1] + sh_x[2] + sh_x[3] + sh_x[4];
    }
}

// Deterministic final reduction of 1280 partials on the matrix unit:
// 20 chunks, each a 16x4 F32 A-tile, times an all-ones 4x16 B, accumulated in
// a 16x16 F32 C. Every column of D then holds the 16 per-row totals.
__global__ __launch_bounds__(32)
void ssim_reduce_kernel(const float* __restrict__ partials,
                        float* __restrict__ out)
{
    const int lane = threadIdx.x;
    const int half = lane >> 4;    // 0: lanes 0-15 (K=0,1) | 1: lanes 16-31 (K=2,3)
    const int m    = lane & 15;    // A-matrix row

    v8f c = {};
    v2f bones;
    bones.x = 1.0f;
    bones.y = 1.0f;

    for (int chunk = 0; chunk < NPARTIAL / 64; ++chunk) {
        const float* p = partials + chunk * 64;   // 16x4 row-major tile
        v2f a;
        a.x = p[m * 4 + 2 * half + 0];   // A[m][2*half]
        a.y = p[m * 4 + 2 * half + 1];   // A[m][2*half+1]
        // (neg_a, A, neg_b, B, c_mod, C, reuse_a, reuse_b)
        c = __builtin_amdgcn_wmma_f32_16x16x4_f32(
                false, a, false, bones, (short)0, c, false, false);
    }

    // Lane<16 holds D rows 0-7 of its column; lane>=16 holds rows 8-15.
    float s = c[0] + c[1] + c[2] + c[3] + c[4] + c[5] + c[6] + c[7];
    float other = __shfl(s, lane ^ 16, 32);
    float total = s + other;

    if (lane == 0) {
        const float n = (float)BATCH * (float)OUT_H * (float)OUT_W;
        out[0] = 1.0f - total / n;
    }
}

extern "C" void kernel_launch(void* const* d_in, const int* in_sizes, int n_in,
                              void* d_out, int out_size, void* d_ws, size_t ws_size,
                              hipStream_t stream)
{
    const float* X  = (const float*)d_in[0];
    const float* Y  = (const float*)d_in[1];
    const float* dr = (const float*)d_in[2];
    // d_in[3] is the uniform 1/49 conv weight; folded into constants.

    float* partials = (float*)d_ws;   // 1280 floats, fully overwritten each call

    dim3 grid(BATCH, NTILES);
    ssim_partial_kernel<<<grid, IMG_W, 0, stream>>>(X, Y, dr, partials);
    ssim_reduce_kernel<<<1, 32, 0, stream>>>(partials, (float*)d_out);
}